// DifferentiableLogicLayer_54674933678729
// MI455X (gfx1250) — compile-verified
//
#include <hip/hip_runtime.h>

typedef __attribute__((ext_vector_type(2))) float v2f;
typedef __attribute__((ext_vector_type(4))) float v4f;
typedef __attribute__((ext_vector_type(8))) float v8f;

#define NUM_GATES   8192
#define INPUT_SIZE  8192
#define BATCH       2048

// Transposed + zero-padded coefficient matrix: CT[n][op], n = coefficient index
// in the basis {1, a, b, ab}, op = soft-logic op index. Rows 4..15 are zero so
// every lane (N column 0..15) can load unconditionally.
__device__ __constant__ float CT[16][16] = {
    //            0   ab  a-ab  a  b-ab  b  xor  or  nor xnor ~b ~b+ab ~a ~a+ab ~ab  1
    /* c0  */ {  0,  0,  0,  0,  0,  0,  0,  0,  1,  1,  1,  1,  1,  1,  1,  1 },
    /* ca  */ {  0,  0,  1,  1,  0,  0,  1,  1, -1, -1,  0,  0, -1, -1,  0,  0 },
    /* cb  */ {  0,  0,  0,  0,  1,  1,  1,  1, -1, -1, -1, -1,  0,  0,  0,  0 },
    /* cab */ {  0,  1, -1,  0, -1,  0, -2, -1,  1,  2,  0,  1,  0,  1, -1,  0 },
    {0},{0},{0},{0},{0},{0},{0},{0},{0},{0},{0},{0}
};

// ---------------------------------------------------------------------------
// Kernel 1: per-gate softmax + W = P @ C via V_WMMA_F32_16X16X4_F32.
// One wave (32 threads) handles a tile of 16 gates.
//   A (16x4 f32): lanes 0-15 -> M=0..15, v0=K0, v1=K1; lanes 16-31 -> v0=K2, v1=K3
//   B (4x16 f32): v0 lanes 0-15 -> K0,N=lane; lanes 16-31 -> K2; v1 -> K1/K3
//   D (16x16 f32): VGPR j, lanes 0-15: M=j, N=lane; lanes 16-31: M=j+8, N=lane-16
// ---------------------------------------------------------------------------
__global__ void __launch_bounds__(32)
gate_weights_wmma(const float* __restrict__ logits, float* __restrict__ w)
{
    __shared__ float sm[16][18];        // 18-pad: &sm[m][even] is 8B-aligned
    const int lane  = threadIdx.x;      // 0..31
    const int g0    = blockIdx.x * 16;  // 16 gates per tile

    // Load 16x16 logits tile (256 contiguous floats).
    const float* src = logits + g0 * 16;
#pragma unroll
    for (int i = 0; i < 8; ++i) {
        int idx = lane * 8 + i;
        sm[idx >> 4][idx & 15] = src[idx];
    }
    __syncthreads();

    const int m     = lane & 15;   // gate row within tile (duplicated across halves)
    const int khalf = lane >> 4;   // 0: K pair {0,1}+4k, 1: K pair {2,3}+4k
    const int n     = m;           // B/D column index for this lane

    // Softmax over the 16 ops of row m (lanes m and m+16 duplicate; cheap).
    float row[16];
    float mx = -INFINITY;
#pragma unroll
    for (int j = 0; j < 16; ++j) { row[j] = sm[m][j]; mx = fmaxf(mx, row[j]); }
    float sum = 0.f;
#pragma unroll
    for (int j = 0; j < 16; ++j) { row[j] = __expf(row[j] - mx); sum += row[j]; }
    const float inv = 1.0f / sum;
#pragma unroll
    for (int j = 0; j < 16; ++j) row[j] *= inv;

    // Publish probabilities so each lane can fetch its K-pair with a plain
    // dynamic LDS load (avoids register-indexing select chains).
    __syncthreads();
    if (lane < 16) {
#pragma unroll
        for (int j = 0; j < 16; ++j) sm[m][j] = row[j];
    }
    __syncthreads();

    // W(16x4) = P(16x16) @ C(16x4), K accumulated over 4 chained 16x16x4 WMMAs.
    v8f acc = {};
#pragma unroll
    for (int k = 0; k < 4; ++k) {
        const int r0 = 4 * k + 2 * khalf;           // even
        const v2f a = *(const v2f*)(&sm[m][r0]);    // ds_load_b64
        const v2f b = *(const v2f*)(&CT[n][r0]);    // b64 from constant, padded
        acc = __builtin_amdgcn_wmma_f32_16x16x4_f32(
                  /*neg_a=*/false, a, /*neg_b=*/false, b,
                  /*c_mod=*/(short)0, acc, /*reuse_a=*/false, /*reuse_b=*/false);
    }

    // Extract D columns 0..3 -> w[g][0..3].
    if (n < 4) {
#pragma unroll
        for (int j = 0; j < 8; ++j) {
            const int grow = j + 8 * khalf;
            w[(g0 + grow) * 4 + n] = acc[j];
        }
    }
}

// ---------------------------------------------------------------------------
// Kernel 2: streaming evaluation. out[b][g] = w0 + wa*a + wb*b + wab*a*b
// Each thread produces 4 consecutive gates of one batch row.
// ---------------------------------------------------------------------------
__global__ void __launch_bounds__(256)
logic_eval(const float* __restrict__ x,
           const v4f* __restrict__ w4,
           float* __restrict__ out)
{
    const int tid = blockIdx.x * blockDim.x + threadIdx.x;
    const int gq  = tid & (NUM_GATES / 4 - 1);   // gate-quad index
    const int bi  = tid >> 11;                   // batch row (NUM_GATES/4 == 2048)
    const int g   = gq << 2;

    const float* xr = x + ((size_t)bi << 13);    // * INPUT_SIZE

    const v4f   xa = *(const v4f*)(xr + g);
    const float xn = xr[(g + 4) & (INPUT_SIZE - 1)];

    float va[5] = { xa.x, xa.y, xa.z, xa.w, xn };
#pragma unroll
    for (int i = 0; i < 5; ++i) va[i] = fminf(fmaxf(va[i], 0.f), 1.f);

    v4f r;
#pragma unroll
    for (int i = 0; i < 4; ++i) {
        const v4f wv = w4[g + i];
        const float a = va[i], b = va[i + 1];
        float t = fmaf(wv.y, a, wv.x);      // w0 + wa*a
        t = fmaf(wv.z, b, t);               // + wb*b
        t = fmaf(wv.w * a, b, t);           // + wab*a*b
        r[i] = t;
    }

    __builtin_nontemporal_store(r, (v4f*)(out + (size_t)tid * 4));
}

extern "C" void kernel_launch(void* const* d_in, const int* in_sizes, int n_in,
                              void* d_out, int out_size, void* d_ws, size_t ws_size,
                              hipStream_t stream)
{
    const float* x      = (const float*)d_in[0];   // (2048, 8192) f32
    const float* logits = (const float*)d_in[1];   // (8192, 16)  f32
    float*       out    = (float*)d_out;           // (2048, 8192) f32
    float*       w      = (float*)d_ws;            // (8192, 4)   f32 scratch

    // 1) Per-gate weights via WMMA: 8192 gates / 16 per tile = 512 waves.
    gate_weights_wmma<<<NUM_GATES / 16, 32, 0, stream>>>(logits, w);

    // 2) Streaming eval: BATCH * NUM_GATES / 4 threads.
    const int total_threads = BATCH * (NUM_GATES / 4);
    logic_eval<<<total_threads / 256, 256, 0, stream>>>(x, (const v4f*)w, out);
}